// SageModel_25280177504627
// MI455X (gfx1250) — compile-verified
//
#include <hip/hip_runtime.h>
#include <hip/hip_bf16.h>

typedef __attribute__((ext_vector_type(2))) float v2f;
typedef __attribute__((ext_vector_type(8))) float v8f;

#define FEATS 256

// ---------------------------------------------------------------------------
// Zero-fill scratch (msg/deg accumulators) — run each call for determinism.
// ---------------------------------------------------------------------------
__global__ void fill_zero(float* __restrict__ p, size_t n) {
    size_t i = (size_t)blockIdx.x * blockDim.x + threadIdx.x;
    if (i < n) p[i] = 0.0f;
}

// ---------------------------------------------------------------------------
// 256x256 transpose via padded LDS tile (32x32 tiles, 256 threads = 32x8).
// Coalesced loads and stores; bank-conflict-free via +1 padding.
// ---------------------------------------------------------------------------
__global__ __launch_bounds__(256)
void transpose256(const float* __restrict__ src, float* __restrict__ dst) {
    __shared__ float tile[32][33];
    const int tx = threadIdx.x & 31;
    const int ty = threadIdx.x >> 5;           // 0..7
    const int bx = blockIdx.x * 32;
    const int by = blockIdx.y * 32;
#pragma unroll
    for (int i = 0; i < 32; i += 8)
        tile[ty + i][tx] = src[(size_t)(by + ty + i) * FEATS + bx + tx];
    __syncthreads();
#pragma unroll
    for (int i = 0; i < 32; i += 8)
        dst[(size_t)(bx + ty + i) * FEATS + by + tx] = tile[tx][ty + i];
}

// ---------------------------------------------------------------------------
// Edge scatter: one wave (32 lanes) per edge. Each lane moves 8 floats
// (2 x float4 loads) and issues 8 fp32 HW atomics (global_atomic_add_f32).
// Lane 0 accumulates the destination degree.
// ---------------------------------------------------------------------------
__global__ __launch_bounds__(256)
void scatter_add(const float* __restrict__ src,
                 const int*   __restrict__ esrc,
                 const int*   __restrict__ edst,
                 float*       __restrict__ msg,
                 float*       __restrict__ deg,
                 int nedges) {
    int e = blockIdx.x * 8 + (threadIdx.x >> 5);
    if (e >= nedges) return;
    int lane = threadIdx.x & 31;
    int s = esrc[e];
    int d = edst[e];
    const float4* srow = (const float4*)(src + (size_t)s * FEATS);
    float* drow = msg + (size_t)d * FEATS;
#pragma unroll
    for (int i = 0; i < 2; ++i) {
        int q = lane + i * 32;          // float4 index within the 256-float row
        float4 v = srow[q];
        float* p = drow + q * 4;
        unsafeAtomicAdd(p + 0, v.x);
        unsafeAtomicAdd(p + 1, v.y);
        unsafeAtomicAdd(p + 2, v.z);
        unsafeAtomicAdd(p + 3, v.w);
    }
    if (lane == 0) unsafeAtomicAdd(deg + d, 1.0f);
}

// ---------------------------------------------------------------------------
// Fused SAGE layer GEMM (weights pre-transposed: Wt[n][k]):
//   out[m,n] = act( Aself[m,:] @ Wself[:,n]
//                 + (Amsg[m,:] / max(deg[m],1)) @ Wneigh[:,n] + bias[n] )
// One wave computes one 16x16 output tile via V_WMMA_F32_16X16X4_F32.
// Block = 128 threads (4 waves) covering 16 rows x 64 cols.
// Grid  = (M/16, FEATS/64).
//
// A-frag (16x4 f32): lane l(<16): {A[m0+l][k],   A[m0+l][k+1]},
//                    lane l+16  : {A[m0+l][k+2], A[m0+l][k+3]}
// B-frag (4x16 f32): lane l(<16): {W[k][n0+l],   W[k+1][n0+l]} = Wt[n0+l][k..k+1]
//                    lane l+16  : {W[k+2][n0+l], W[k+3][n0+l]} = Wt[n0+l][k+2..k+3]
// --> both fragments are single aligned float2 global loads.
// C/D layout: VGPR v -> M = m0 + v + 8*half, N = n0 + (lane&15)
// ---------------------------------------------------------------------------
__global__ __launch_bounds__(128)
void sage_gemm(const float* __restrict__ Aself,
               const float* __restrict__ Amsg,
               const float* __restrict__ deg,
               const float* __restrict__ WselfT,
               const float* __restrict__ WneighT,
               const float* __restrict__ bias,
               float*       __restrict__ out,
               int relu) {
    const int lane = threadIdx.x & 31;
    const int wave = threadIdx.x >> 5;
    const int half = lane >> 4;       // 0 or 1
    const int l    = lane & 15;
    const int m0   = blockIdx.x * 16;
    const int n0   = (blockIdx.y * 4 + wave) * 16;
    const int row  = m0 + l;
    const int col  = n0 + l;

    v8f acc = {};

    const float* arow  = Aself  + (size_t)row * FEATS + half * 2;
    const float* wsrow = WselfT + (size_t)col * FEATS + half * 2;

    // ---- self term: Aself @ Wself --------------------------------------
#pragma unroll 8
    for (int k = 0; k < FEATS; k += 4) {
        float2 av = *(const float2*)(arow + k);      // 8B aligned
        float2 bv = *(const float2*)(wsrow + k);
        v2f a; a.x = av.x; a.y = av.y;
        v2f b; b.x = bv.x; b.y = bv.y;
        acc = __builtin_amdgcn_wmma_f32_16x16x4_f32(
                  false, a, false, b, (short)0, acc, false, false);
    }

    // ---- neighbor term: (Amsg/deg) @ Wneigh ----------------------------
    const float invd = 1.0f / fmaxf(deg[row], 1.0f);
    const float* mrow  = Amsg    + (size_t)row * FEATS + half * 2;
    const float* wnrow = WneighT + (size_t)col * FEATS + half * 2;
#pragma unroll 8
    for (int k = 0; k < FEATS; k += 4) {
        float2 av = *(const float2*)(mrow + k);
        float2 bv = *(const float2*)(wnrow + k);
        v2f a; a.x = av.x * invd; a.y = av.y * invd;
        v2f b; b.x = bv.x; b.y = bv.y;
        acc = __builtin_amdgcn_wmma_f32_16x16x4_f32(
                  false, a, false, b, (short)0, acc, false, false);
    }

    // ---- bias + activation + store -------------------------------------
    const float bb = bias[col];
#pragma unroll
    for (int v = 0; v < 8; ++v) {
        int mo = m0 + v + half * 8;
        float r = acc[v] + bb;
        if (relu) r = fmaxf(r, 0.0f);
        out[(size_t)mo * FEATS + col] = r;
    }
}

// ---------------------------------------------------------------------------
// Host-side orchestration
// ---------------------------------------------------------------------------
#define N0 131072
#define N1 32768
#define N2 8192
#define E0 524288
#define E1 131072

extern "C" void kernel_launch(void* const* d_in, const int* in_sizes, int n_in,
                              void* d_out, int out_size, void* d_ws, size_t ws_size,
                              hipStream_t stream) {
    (void)in_sizes; (void)n_in; (void)out_size; (void)ws_size;

    const float* x       = (const float*)d_in[0];
    const float* Wself1  = (const float*)d_in[1];
    const float* Wneigh1 = (const float*)d_in[2];
    const float* b1      = (const float*)d_in[3];
    const float* Wself2  = (const float*)d_in[4];
    const float* Wneigh2 = (const float*)d_in[5];
    const float* b2      = (const float*)d_in[6];
    const int*   e0s     = (const int*)d_in[7];
    const int*   e0d     = (const int*)d_in[8];
    const int*   e1s     = (const int*)d_in[9];
    const int*   e1d     = (const int*)d_in[10];

    char* ws = (char*)d_ws;
    const size_t msg1_b = (size_t)N1 * FEATS * 4;   // 33554432
    const size_t deg1_b = (size_t)N1 * 4;           //   131072
    const size_t msg2_b = (size_t)N2 * FEATS * 4;   //  8388608
    const size_t deg2_b = (size_t)N2 * 4;           //    32768
    const size_t h1_b   = (size_t)N1 * FEATS * 4;   // 33554432
    const size_t wt_b   = (size_t)FEATS * FEATS * 4;//   262144
    size_t off = 0;
    float* msg1 = (float*)(ws + off); off += msg1_b;
    float* deg1 = (float*)(ws + off); off += deg1_b;
    float* msg2 = (float*)(ws + off); off += msg2_b;
    float* deg2 = (float*)(ws + off); off += deg2_b;
    float* h1   = (float*)(ws + off); off += h1_b;
    float* Wt1s = (float*)(ws + off); off += wt_b;
    float* Wt1n = (float*)(ws + off); off += wt_b;
    float* Wt2s = (float*)(ws + off); off += wt_b;
    float* Wt2n = (float*)(ws + off); off += wt_b;

    // Zero all accumulators (msg1|deg1|msg2|deg2 are contiguous).
    size_t nz = (msg1_b + deg1_b + msg2_b + deg2_b) / 4;
    fill_zero<<<(unsigned)((nz + 255) / 256), 256, 0, stream>>>((float*)ws, nz);

    // Pre-transpose all four weight matrices (K-major -> N-major).
    dim3 tg(FEATS / 32, FEATS / 32);
    transpose256<<<tg, 256, 0, stream>>>(Wself1,  Wt1s);
    transpose256<<<tg, 256, 0, stream>>>(Wneigh1, Wt1n);
    transpose256<<<tg, 256, 0, stream>>>(Wself2,  Wt2s);
    transpose256<<<tg, 256, 0, stream>>>(Wneigh2, Wt2n);

    // Layer 1: aggregate then fused GEMM (+bias, ReLU) -> h1 (32768 x 256)
    scatter_add<<<E0 / 8, 256, 0, stream>>>(x, e0s, e0d, msg1, deg1, E0);
    sage_gemm<<<dim3(N1 / 16, FEATS / 64), 128, 0, stream>>>(
        x, msg1, deg1, Wt1s, Wt1n, b1, h1, 1);

    // Layer 2: aggregate then fused GEMM (+bias) -> d_out (8192 x 256)
    scatter_add<<<E1 / 8, 256, 0, stream>>>(h1, e1s, e1d, msg2, deg2, E1);
    sage_gemm<<<dim3(N2 / 16, FEATS / 64), 128, 0, stream>>>(
        h1, msg2, deg2, Wt2s, Wt2n, b2, (float*)d_out, 0);
}